// S4Model_50087908606106
// MI455X (gfx1250) — compile-verified
//
#include <hip/hip_runtime.h>
#include <hip/hip_bf16.h>
#include <math.h>

// ---------------------------------------------------------------------------
// S4D model forward for MI455X (gfx1250, wave32).
//   encoder: 3x (conv1d SAME + BN(eval) + ReLU)
//   4x S4D layer: Vandermonde kernel -> 8192-pt LDS FFT conv -> D skip -> GELU
//                 -> WMMA f32 GEMM (H=256 -> 2H=512, Z stored [B,L,512])
//                 -> GLU -> residual -> LN(ch)
//   mean-pool over time -> linear decode (10 classes)
// ---------------------------------------------------------------------------

#define BATCH 16
#define LSEQ  4096
#define HDIM  256
#define NSTATE 64
#define NLAYER 4
#define ODIM  512      // 2*H
#define NFFT  8192
#define LOG2_NFFT 13
#define DOUT  10

typedef float v2f __attribute__((ext_vector_type(2)));
typedef float v8f __attribute__((ext_vector_type(8)));

// ---------------------------------------------------------------------------
// CDNA5 async global->LDS copy (per-lane LDS destination = scatter capable).
// VDST vgpr carries the LDS byte address; tracked by ASYNCcnt.
// ---------------------------------------------------------------------------
__device__ __forceinline__ void async_g2l_b32(const float* gptr, float* lptr) {
    unsigned lds = (unsigned)(uintptr_t)lptr;              // low 32 bits = LDS addr
    unsigned long long ga = (unsigned long long)(uintptr_t)gptr;
    asm volatile("global_load_async_to_lds_b32 %0, %1, off"
                 :: "v"(lds), "v"(ga) : "memory");
}
__device__ __forceinline__ void async_wait0() {
    asm volatile("s_wait_asynccnt 0x0" ::: "memory");
}

// ---------------------------------------------------------------------------
// Encoder: direct conv1d (SAME, stride 1, odd K) fused with BN(eval) + ReLU.
// grid: (L/256, Co, B), block: 256, dyn LDS: Ci*K floats (weights for this co)
// ---------------------------------------------------------------------------
__global__ void conv_bn_relu_kernel(const float* __restrict__ X,   // [B,Ci,L]
                                    const float* __restrict__ W,   // [Co,Ci,K]
                                    const float* __restrict__ cb,  // [Co]
                                    const float* __restrict__ bn_g,
                                    const float* __restrict__ bn_b,
                                    const float* __restrict__ bn_m,
                                    const float* __restrict__ bn_v,
                                    float* __restrict__ Y,         // [B,Co,L]
                                    int Ci, int K, int Co) {
    extern __shared__ float wsm[];
    const int co = blockIdx.y;
    const int b  = blockIdx.z;
    const int l  = blockIdx.x * blockDim.x + threadIdx.x;
    for (int i = threadIdx.x; i < Ci * K; i += blockDim.x)
        wsm[i] = W[(size_t)co * Ci * K + i];
    __syncthreads();
    const int pad = K >> 1;
    const float* xb = X + (size_t)b * Ci * LSEQ;
    float acc = 0.0f;
    for (int ci = 0; ci < Ci; ++ci) {
        const float* xr = xb + (size_t)ci * LSEQ;
        const float* wr = wsm + ci * K;
        #pragma unroll 7
        for (int k = 0; k < K; ++k) {
            int pos = l + k - pad;
            float xv = (pos >= 0 && pos < LSEQ) ? xr[pos] : 0.0f;
            acc = fmaf(wr[k], xv, acc);
        }
    }
    float s = bn_g[co] * rsqrtf(bn_v[co] + 1e-5f);
    float t = bn_b[co] - bn_m[co] * s;
    float y = fmaf(acc + cb[co], s, t);
    Y[((size_t)b * Co + co) * LSEQ + l] = fmaxf(y, 0.0f);
}

// ---------------------------------------------------------------------------
// S4D kernel generation: K[h,l] = 2*Re sum_n coef[h,n] * exp(dtA[h,n]*l)
// grid: H blocks, 256 threads.
// ---------------------------------------------------------------------------
__global__ void s4_gen_kernel(const float* __restrict__ log_dt,   // [H]
                              const float* __restrict__ A_re_log, // [H,N]
                              const float* __restrict__ A_im,     // [H,N]
                              const float* __restrict__ C_re,     // [H,N]
                              const float* __restrict__ C_im,     // [H,N]
                              float* __restrict__ Kt)             // [H,L]
{
    __shared__ float sa[NSTATE], sb[NSTATE], scr[NSTATE], sci[NSTATE];
    const int h = blockIdx.x;
    const int t = threadIdx.x;
    if (t < NSTATE) {
        float dt  = __expf(log_dt[h]);
        float Are = -__expf(A_re_log[h * NSTATE + t]);
        float Aim = A_im[h * NSTATE + t];
        float ar = Are * dt, ai = Aim * dt;            // dtA
        float er = __expf(ar);
        float sn, cs; __sincosf(ai, &sn, &cs);
        float Er = er * cs - 1.0f, Ei = er * sn;       // exp(dtA)-1
        float Cr = C_re[h * NSTATE + t], Ci = C_im[h * NSTATE + t];
        float nr = Cr * Er - Ci * Ei, ni = Cr * Ei + Ci * Er;  // C*(exp(dtA)-1)
        float d2 = Are * Are + Aim * Aim;
        sa[t]  = ar;  sb[t]  = ai;
        scr[t] = (nr * Are + ni * Aim) / d2;           // coef = num / A
        sci[t] = (ni * Are - nr * Aim) / d2;
    }
    __syncthreads();
    for (int l = t; l < LSEQ; l += blockDim.x) {
        float fl = (float)l;
        float acc = 0.0f;
        #pragma unroll 4
        for (int n = 0; n < NSTATE; ++n) {
            float e = __expf(sa[n] * fl);
            float sn, cs; __sincosf(sb[n] * fl, &sn, &cs);
            acc = fmaf(e, scr[n] * cs - sci[n] * sn, acc);
        }
        Kt[(size_t)h * LSEQ + l] = 2.0f * acc;
    }
}

// ---------------------------------------------------------------------------
// In-LDS radix-2 DIT FFT, N=8192 complex (64KB LDS), blockDim=256, wave32.
// dir = -1 forward, +1 inverse (no 1/N scaling inside).
// ---------------------------------------------------------------------------
__device__ __forceinline__ void fft8192_lds(float* re, float* im, float dir) {
    const int t = threadIdx.x;
    for (int len = 2; len <= NFFT; len <<= 1) {
        const int half = len >> 1;
        const float ang0 = dir * 6.28318530717958647692f / (float)len;
        for (int i = t; i < (NFFT >> 1); i += 256) {
            int j = i & (half - 1);
            int pos = ((i - j) << 1) + j;
            float sn, cs; __sincosf(ang0 * (float)j, &sn, &cs);
            float xr = re[pos + half], xi = im[pos + half];
            float tr = xr * cs - xi * sn;
            float ti = xr * sn + xi * cs;
            re[pos + half] = re[pos] - tr;
            im[pos + half] = im[pos] - ti;
            re[pos] += tr;
            im[pos] += ti;
        }
        __syncthreads();
    }
}

__device__ __forceinline__ int bitrev13(int i) { return (int)(__brev((unsigned)i) >> (32 - LOG2_NFFT)); }

// Forward FFT of the SSM kernel K: Kt[h,0:L] zero-padded to 8192 -> Kf[h] full spectrum.
// grid: H blocks, 256 threads, dyn LDS 64KB.
__global__ void fftK_kernel(const float* __restrict__ Kt, float* __restrict__ Kf) {
    extern __shared__ float sm[];
    float* re = sm;
    float* im = sm + NFFT;
    const int h = blockIdx.x;
    const int t = threadIdx.x;
    for (int i = t; i < LSEQ; i += 256)                  // async bit-reversed scatter
        async_g2l_b32(Kt + (size_t)h * LSEQ + i, re + bitrev13(i));
    for (int i = LSEQ + t; i < NFFT; i += 256)           // zero-pad tail
        re[bitrev13(i)] = 0.0f;
    for (int i = t; i < NFFT; i += 256)
        im[i] = 0.0f;
    async_wait0();
    __syncthreads();
    fft8192_lds(re, im, -1.0f);
    for (int i = t; i < NFFT; i += 256) {
        Kf[((size_t)h * NFFT + i) * 2 + 0] = re[i];
        Kf[((size_t)h * NFFT + i) * 2 + 1] = im[i];
    }
}

// ---------------------------------------------------------------------------
// FFT convolution per (b,h) sequence: y = irfft(fft(u)*Kf)[0:L] + D*u, GELU.
// grid: B*H blocks, 256 threads, dyn LDS 64KB.
// ---------------------------------------------------------------------------
__global__ void s4_fftconv_kernel(const float* __restrict__ X,  // [B,H,L]
                                  const float* __restrict__ Kf, // [H,8192,2]
                                  const float* __restrict__ D,  // [H]
                                  float* __restrict__ Y)        // [B,H,L]
{
    extern __shared__ float sm[];
    float* re = sm;
    float* im = sm + NFFT;
    const int bh = blockIdx.x;
    const int h  = bh & (HDIM - 1);
    const float* u = X + (size_t)bh * LSEQ;
    const int t = threadIdx.x;

    for (int i = t; i < LSEQ; i += 256)                  // async bit-reversed scatter
        async_g2l_b32(u + i, re + bitrev13(i));
    for (int i = LSEQ + t; i < NFFT; i += 256)
        re[bitrev13(i)] = 0.0f;
    for (int i = t; i < NFFT; i += 256)
        im[i] = 0.0f;
    async_wait0();
    __syncthreads();
    fft8192_lds(re, im, -1.0f);                       // U(f), natural order

    const float* kf = Kf + (size_t)h * NFFT * 2;
    for (int i = t; i < NFFT; i += 256) {             // pointwise U*K
        float ur = re[i], ui = im[i];
        float kr = kf[2 * i], ki = kf[2 * i + 1];
        re[i] = ur * kr - ui * ki;
        im[i] = ur * ki + ui * kr;
    }
    __syncthreads();
    for (int i = t; i < NFFT; i += 256) {             // bit-reverse permute in place
        int r = bitrev13(i);
        if (r > i) {
            float a = re[i]; re[i] = re[r]; re[r] = a;
            float b = im[i]; im[i] = im[r]; im[r] = b;
        }
    }
    __syncthreads();
    fft8192_lds(re, im, +1.0f);                       // inverse

    const float Dh = D[h];
    const float inv_n = 1.0f / (float)NFFT;
    for (int l = t; l < LSEQ; l += 256) {
        float y = fmaf(re[l], inv_n, Dh * u[l]);      // conv + skip
        y = 0.5f * y * (1.0f + erff(y * 0.70710678118654752f));  // exact GELU
        Y[(size_t)bh * LSEQ + l] = y;
    }
}

// ---------------------------------------------------------------------------
// Position-wise output linear via f32 WMMA: Zt[b,l,:] = W(512x256) @ Y[b](256xL) + bias
// One wave computes a 16(M) x 64(N) strip: 4 accumulators share each A fragment
// (4x A-traffic reduction, 4 back-to-back V_WMMA_F32_16X16X4_F32 per k-step).
// Z is stored transposed [B, L, 512]: the D-fragment layout gives each lane 8
// consecutive M values -> contiguous 32B stores, and the GLU/LN kernel reads
// channels fully coalesced.
// grid: B*32*64/8 = 4096 blocks of 256 threads (8 waves).
// A frag (16x4 f32): lanes 0-15 -> M=lane, K={k,k+1}; lanes 16-31 -> K={k+2,k+3}.
// B frag (4x16 f32): lanes 0-15 -> N=lane, K={k,k+1}; lanes 16-31 -> K={k+2,k+3}.
// D: VGPR r -> M = r + 8*(lane>=16), N = lane&15.
// ---------------------------------------------------------------------------
__global__ void out_linear_wmma_kernel(const float* __restrict__ W,    // [512,256]
                                       const float* __restrict__ bias, // [512]
                                       const float* __restrict__ Y,    // [B,256,L]
                                       float* __restrict__ Zt)         // [B,L,512]
{
    const int groupsN = LSEQ / 64;         // 64 n-groups of 4 tiles
    const int tilesM  = ODIM / 16;         // 32
    const int perBatch = tilesM * groupsN; // 2048 waves per batch

    const int wave = blockIdx.x * (blockDim.x >> 5) + (threadIdx.x >> 5);
    const int lane = threadIdx.x & 31;
    const int b  = wave / perBatch;
    const int r0 = wave % perBatch;
    const int tm = r0 / groupsN;
    const int tg = r0 % groupsN;
    const int m0 = tm * 16;
    const int n0 = tg * 64;
    const int laneL = lane & 15;
    const int hi    = lane >> 4;           // 0 or 1

    const float* __restrict__ Arow = W + (size_t)(m0 + laneL) * HDIM;  // A: row m
    const float* __restrict__ Bbase = Y + (size_t)b * HDIM * LSEQ + laneL;

    v8f acc0 = {}, acc1 = {}, acc2 = {}, acc3 = {};
    #pragma unroll 4
    for (int k = 0; k < HDIM; k += 4) {
        const int ka = k + 2 * hi;
        v2f a;
        a.x = Arow[ka];
        a.y = Arow[ka + 1];
        __builtin_prefetch(Arow + k + 8, 0, 3);                // global_prefetch_b8
        const float* brow0 = Bbase + (size_t)ka * LSEQ + n0;
        const float* brow1 = brow0 + LSEQ;
        v2f b0; b0.x = brow0[0];  b0.y = brow1[0];
        v2f b1; b1.x = brow0[16]; b1.y = brow1[16];
        v2f b2; b2.x = brow0[32]; b2.y = brow1[32];
        v2f b3; b3.x = brow0[48]; b3.y = brow1[48];
        acc0 = __builtin_amdgcn_wmma_f32_16x16x4_f32(false, a, false, b0, (short)0, acc0, false, false);
        acc1 = __builtin_amdgcn_wmma_f32_16x16x4_f32(false, a, false, b1, (short)0, acc1, false, false);
        acc2 = __builtin_amdgcn_wmma_f32_16x16x4_f32(false, a, false, b2, (short)0, acc2, false, false);
        acc3 = __builtin_amdgcn_wmma_f32_16x16x4_f32(false, a, false, b3, (short)0, acc3, false, false);
    }

    // epilogue: each lane owns 8 consecutive channels [m0+8*hi, m0+8*hi+8)
    float bsum[8];
    #pragma unroll
    for (int r = 0; r < 8; ++r) bsum[r] = bias[m0 + 8 * hi + r];

    float* __restrict__ zb = Zt + ((size_t)b * LSEQ) * ODIM + m0 + 8 * hi;
    v8f* accs[4] = { &acc0, &acc1, &acc2, &acc3 };
    #pragma unroll
    for (int j = 0; j < 4; ++j) {
        float* zp = zb + (size_t)(n0 + 16 * j + laneL) * ODIM;
        #pragma unroll
        for (int r = 0; r < 8; ++r)
            zp[r] = (*accs[j])[r] + bsum[r];
    }
}

// ---------------------------------------------------------------------------
// GLU + residual + channel LayerNorm: x = LN_ch(z1*sigmoid(z2) + x)
// Z is [B,L,512] -> channel reads are fully coalesced.
// grid: (L, B), block: 256 (one thread per channel).
// ---------------------------------------------------------------------------
__global__ void glu_res_ln_kernel(const float* __restrict__ Zt, // [B,L,512]
                                  float* __restrict__ X,        // [B,256,L] in/out
                                  const float* __restrict__ g,  // [256]
                                  const float* __restrict__ bt) // [256]
{
    __shared__ float red[HDIM];
    const int l = blockIdx.x;
    const int b = blockIdx.y;
    const int h = threadIdx.x;
    const float* zrow = Zt + ((size_t)b * LSEQ + l) * ODIM;
    float z1 = zrow[h];
    float z2 = zrow[h + HDIM];
    float v  = z1 / (1.0f + __expf(-z2));              // GLU
    const size_t xi = ((size_t)b * HDIM + h) * LSEQ + l;
    v += X[xi];                                        // residual

    red[h] = v; __syncthreads();
    for (int s = HDIM / 2; s > 0; s >>= 1) {
        if (h < s) red[h] += red[h + s];
        __syncthreads();
    }
    const float mu = red[0] * (1.0f / HDIM);
    __syncthreads();
    float d = v - mu;
    red[h] = d * d; __syncthreads();
    for (int s = HDIM / 2; s > 0; s >>= 1) {
        if (h < s) red[h] += red[h + s];
        __syncthreads();
    }
    const float var = red[0] * (1.0f / HDIM);
    X[xi] = d * rsqrtf(var + 1e-5f) * g[h] + bt[h];
}

// ---------------------------------------------------------------------------
// Mean-pool over time. grid: B*H blocks, 256 threads.
// ---------------------------------------------------------------------------
__global__ void pool_kernel(const float* __restrict__ X, float* __restrict__ P) {
    __shared__ float red[256];
    const int bh = blockIdx.x;
    const float* xr = X + (size_t)bh * LSEQ;
    float acc = 0.0f;
    for (int i = threadIdx.x; i < LSEQ; i += 256) acc += xr[i];
    red[threadIdx.x] = acc; __syncthreads();
    for (int s = 128; s > 0; s >>= 1) {
        if (threadIdx.x < s) red[threadIdx.x] += red[threadIdx.x + s];
        __syncthreads();
    }
    if (threadIdx.x == 0) P[bh] = red[0] * (1.0f / LSEQ);
}

// Decoder: out[b,d] = sum_h P[b,h]*dec_w[d,h] + dec_b[d]. grid: B blocks.
__global__ void decode_kernel(const float* __restrict__ P,
                              const float* __restrict__ dw,
                              const float* __restrict__ db,
                              float* __restrict__ out) {
    __shared__ float red[HDIM];
    const int b = blockIdx.x;
    const int h = threadIdx.x;
    const float p = P[b * HDIM + h];
    for (int d = 0; d < DOUT; ++d) {
        red[h] = p * dw[d * HDIM + h];
        __syncthreads();
        for (int s = HDIM / 2; s > 0; s >>= 1) {
            if (h < s) red[h] += red[h + s];
            __syncthreads();
        }
        if (h == 0) out[b * DOUT + d] = red[0] + db[d];
        __syncthreads();
    }
}

// ---------------------------------------------------------------------------
extern "C" void kernel_launch(void* const* d_in, const int* in_sizes, int n_in,
                              void* d_out, int out_size, void* d_ws, size_t ws_size,
                              hipStream_t stream) {
    (void)in_sizes; (void)n_in; (void)out_size; (void)ws_size;

    const float* x        = (const float*)d_in[0];
    const float* conv_w[3] = { (const float*)d_in[1],  (const float*)d_in[7],  (const float*)d_in[13] };
    const float* conv_b[3] = { (const float*)d_in[2],  (const float*)d_in[8],  (const float*)d_in[14] };
    const float* bn_g[3]   = { (const float*)d_in[3],  (const float*)d_in[9],  (const float*)d_in[15] };
    const float* bn_b[3]   = { (const float*)d_in[4],  (const float*)d_in[10], (const float*)d_in[16] };
    const float* bn_m[3]   = { (const float*)d_in[5],  (const float*)d_in[11], (const float*)d_in[17] };
    const float* bn_v[3]   = { (const float*)d_in[6],  (const float*)d_in[12], (const float*)d_in[18] };
    const float* log_dt   = (const float*)d_in[19];   // [NL,H]
    const float* A_re_log = (const float*)d_in[20];   // [NL,H,N]
    const float* A_im     = (const float*)d_in[21];
    const float* C_re     = (const float*)d_in[22];
    const float* C_im     = (const float*)d_in[23];
    const float* Dp       = (const float*)d_in[24];   // [NL,H]
    const float* out_w    = (const float*)d_in[25];   // [NL,512,256]
    const float* out_b    = (const float*)d_in[26];   // [NL,512]
    const float* ln_g     = (const float*)d_in[27];   // [NL,H]
    const float* ln_b     = (const float*)d_in[28];
    const float* dec_w    = (const float*)d_in[29];   // [10,H]
    const float* dec_b    = (const float*)d_in[30];
    float* out            = (float*)d_out;

    // workspace layout (floats)
    float* base = (float*)d_ws;
    size_t off = 0;
    float* E0 = base + off; off += (size_t)BATCH * 64  * LSEQ;        // enc stage 0
    float* E1 = base + off; off += (size_t)BATCH * 128 * LSEQ;        // enc stage 1
    float* Xb = base + off; off += (size_t)BATCH * HDIM * LSEQ;       // activation
    float* Yb = base + off; off += (size_t)BATCH * HDIM * LSEQ;       // s4 gelu out
    float* Zb = base + off; off += (size_t)BATCH * (size_t)LSEQ * ODIM; // linear out [B,L,512]
    float* Kt = base + off; off += (size_t)HDIM * LSEQ;               // time-domain K
    float* Kf = base + off; off += (size_t)HDIM * NFFT * 2;           // spectrum of K
    float* P  = base + off; off += (size_t)BATCH * HDIM;              // pooled

    const dim3 blk(256);
    const size_t fft_lds = (size_t)NFFT * 2 * sizeof(float);          // 64 KB

    // ---- encoder ----
    conv_bn_relu_kernel<<<dim3(LSEQ / 256, 64, BATCH), blk, 1 * 7 * sizeof(float), stream>>>(
        x, conv_w[0], conv_b[0], bn_g[0], bn_b[0], bn_m[0], bn_v[0], E0, 1, 7, 64);
    conv_bn_relu_kernel<<<dim3(LSEQ / 256, 128, BATCH), blk, 64 * 5 * sizeof(float), stream>>>(
        E0, conv_w[1], conv_b[1], bn_g[1], bn_b[1], bn_m[1], bn_v[1], E1, 64, 5, 128);
    conv_bn_relu_kernel<<<dim3(LSEQ / 256, 256, BATCH), blk, 128 * 3 * sizeof(float), stream>>>(
        E1, conv_w[2], conv_b[2], bn_g[2], bn_b[2], bn_m[2], bn_v[2], Xb, 128, 3, 256);

    // ---- S4D layers ----
    const int gemm_waves  = BATCH * (ODIM / 16) * (LSEQ / 64);        // 32768
    const int gemm_blocks = gemm_waves / 8;                           // 4096
    for (int l = 0; l < NLAYER; ++l) {
        const size_t hn = (size_t)l * HDIM * NSTATE;
        s4_gen_kernel<<<HDIM, blk, 0, stream>>>(
            log_dt + (size_t)l * HDIM, A_re_log + hn, A_im + hn, C_re + hn, C_im + hn, Kt);
        fftK_kernel<<<HDIM, blk, fft_lds, stream>>>(Kt, Kf);
        s4_fftconv_kernel<<<BATCH * HDIM, blk, fft_lds, stream>>>(
            Xb, Kf, Dp + (size_t)l * HDIM, Yb);
        out_linear_wmma_kernel<<<gemm_blocks, blk, 0, stream>>>(
            out_w + (size_t)l * ODIM * HDIM, out_b + (size_t)l * ODIM, Yb, Zb);
        glu_res_ln_kernel<<<dim3(LSEQ, BATCH), blk, 0, stream>>>(
            Zb, Xb, ln_g + (size_t)l * HDIM, ln_b + (size_t)l * HDIM);
    }

    // ---- head ----
    pool_kernel<<<BATCH * HDIM, blk, 0, stream>>>(Xb, P);
    decode_kernel<<<BATCH, blk, 0, stream>>>(P, dec_w, dec_b, out);
}